// SGatingUnit_19610820673581
// MI455X (gfx1250) — compile-verified
//
#include <hip/hip_runtime.h>

// SGatingUnit fused kernel for MI455X (gfx1250, wave32).
// One workgroup (512 threads = 16 waves) per window w:
//   Phase 0: load u-half of x once, LayerNorm rows (b,n), scatter normalized
//            values into LDS in V_WMMA_F32_16X16X4_F32 B-operand layout.
//   Head loop (s=0..15): analytic Gaussian scores + softmax into LDS in
//            A-operand layout; 16 waves each compute one 16x16 tile of the
//            64x64 (m x (b,v)) product via 16 chained v_wmma_f32_16x16x4_f32;
//            result overwrites the (now dead) B region of head s as mixed[s].
//   Epilogue: out = (mixed + bias[w,n]) * v   (v streamed from x, coalesced).
// Global traffic per block: 512KB read (x) + 256KB write (out) = HBM minimum.

typedef __attribute__((ext_vector_type(2))) float v2f;
typedef __attribute__((ext_vector_type(8))) float v8f;

#define NWIN   1024
#define NGAMMA 16
#define NTOK   64
#define GATE   256
#define NBATCH 4
#define REG_STRIDE 1032                 // dwords per (head,batch) region: 1024 + pad
#define WWA_BASE   (NGAMMA * NBATCH * REG_STRIDE)   // 66048 dwords
#define LDS_FLOATS (WWA_BASE + 64 * 64)             // + 16KB A-operand buffer

__global__ __launch_bounds__(512) void sgating_fused_kernel(
    const float* __restrict__ x,        // (4, 1024, 64, 512)
    const float* __restrict__ nw,       // (256,)
    const float* __restrict__ nb,       // (256,)
    const float* __restrict__ centers,  // (16384, 2)
    const float* __restrict__ spreads,  // (16384,)
    const float* __restrict__ tbias,    // (1024, 64, 1)
    float* __restrict__ out)            // (4, 1024, 64, 256)
{
    extern __shared__ float lds[];
    const int tid  = threadIdx.x;
    const int lane = tid & 31;
    const int wv   = tid >> 5;          // wave id 0..15
    const int w    = blockIdx.x;        // window

    // ---- hoisted per-lane LayerNorm affine params (channels lane*4.. and +128) ----
    const float4 nw0 = *(const float4*)(nw + lane * 4);
    const float4 nw1 = *(const float4*)(nw + 128 + lane * 4);
    const float4 nb0 = *(const float4*)(nb + lane * 4);
    const float4 nb1 = *(const float4*)(nb + 128 + lane * 4);

    // ================= Phase 0: LayerNorm u and scatter to B-operand layout ======
    // B element mapping for head s, batch b, token n, channel c = v*16+s:
    //   region = (s*4+b)*REG_STRIDE
    //   idx    = region + (kc*32 + (v + 16*((n&2)>>1)))*2 + (n&1),  kc = n>>2
    for (int i = 0; i < 16; ++i) {
        const int row = wv * 16 + i;            // 0..255 = (b,n)
        const int b   = row >> 6;
        const int n   = row & 63;
        const float* urow = x + (((size_t)b * NWIN + w) * NTOK + n) * (2 * GATE);
        const float4 u0 = *(const float4*)(urow + lane * 4);
        const float4 u1 = *(const float4*)(urow + 128 + lane * 4);

        float s1 = u0.x + u0.y + u0.z + u0.w + u1.x + u1.y + u1.z + u1.w;
        float s2 = u0.x*u0.x + u0.y*u0.y + u0.z*u0.z + u0.w*u0.w
                 + u1.x*u1.x + u1.y*u1.y + u1.z*u1.z + u1.w*u1.w;
        #pragma unroll
        for (int d = 16; d >= 1; d >>= 1) {
            s1 += __shfl_xor(s1, d);
            s2 += __shfl_xor(s2, d);
        }
        const float mean = s1 * (1.0f / 256.0f);
        const float var  = s2 * (1.0f / 256.0f) - mean * mean;
        const float rstd = rsqrtf(var + 1e-5f);

        const int kc    = n >> 2;
        const int lwofs = (n & 2) << 3;         // +16 lane offset for k%4 in {2,3}
        const int r     = n & 1;
        auto scat = [&](int c, float uval, float gw, float gb) {
            const float val = (uval - mean) * rstd * gw + gb;
            const int sH = c & 15, vv = c >> 4;
            const int idx = (sH * NBATCH + b) * REG_STRIDE
                          + ((kc * 32 + (vv + lwofs)) << 1) + r;
            lds[idx] = val;
        };
        scat(lane*4 + 0,       u0.x, nw0.x, nb0.x);
        scat(lane*4 + 1,       u0.y, nw0.y, nb0.y);
        scat(lane*4 + 2,       u0.z, nw0.z, nb0.z);
        scat(lane*4 + 3,       u0.w, nw0.w, nb0.w);
        scat(128 + lane*4 + 0, u1.x, nw1.x, nb1.x);
        scat(128 + lane*4 + 1, u1.y, nw1.y, nb1.y);
        scat(128 + lane*4 + 2, u1.z, nw1.z, nb1.z);
        scat(128 + lane*4 + 3, u1.w, nw1.w, nb1.w);
    }
    __syncthreads();

    // softmax thread mapping: 8 threads per row m, 8 cols each
    const int m    = tid >> 3;
    const int cg   = tid & 7;
    const int mt_s = m >> 4;
    const int ml   = m & 15;
    // matmul wave mapping: tile (mt2 rows, batch bb)
    const int mt2 = wv >> 2;
    const int bb  = wv & 3;

    // ================= Head loop =================================================
    for (int s = 0; s < NGAMMA; ++s) {
        // --- Gaussian scores + softmax (analytic rel indices) into A layout ---
        const size_t hr = (size_t)s * NWIN + w;
        const float mu1 = centers[2 * hr];
        const float mu2 = centers[2 * hr + 1];
        float aa = spreads[hr];
        aa = aa * aa;
        const float p0 = aa * mu1, p1 = aa * mu2;

        float sc[8];
        float mx = -3.4e38f;
        #pragma unroll
        for (int i = 0; i < 8; ++i) {
            const int n = cg * 8 + i;
            const float dx = (float)((n & 7) - (m & 7));
            const float dy = (float)((m >> 3) - (n >> 3));
            const float v = p0 * dx + p1 * dy - 0.5f * (dx * dx + dy * dy);
            sc[i] = v;
            mx = fmaxf(mx, v);
        }
        #pragma unroll
        for (int d = 1; d < 8; d <<= 1) mx = fmaxf(mx, __shfl_xor(mx, d));
        float sum = 0.0f;
        #pragma unroll
        for (int i = 0; i < 8; ++i) { sc[i] = __expf(sc[i] - mx); sum += sc[i]; }
        #pragma unroll
        for (int d = 1; d < 8; d <<= 1) sum += __shfl_xor(sum, d);
        const float inv = 1.0f / sum;

        #pragma unroll
        for (int p = 0; p < 4; ++p) {
            const int n0 = cg * 8 + p * 2;      // even
            const int kc2 = n0 >> 2;
            const int lw  = ml + ((n0 & 2) << 3);
            v2f pr;
            pr.x = sc[2 * p] * inv;
            pr.y = sc[2 * p + 1] * inv;
            *(v2f*)&lds[WWA_BASE + (((mt_s * 16 + kc2) * 32 + lw) << 1)] = pr;
        }
        __syncthreads();

        // --- 64x64x64 product: one 16x16 tile per wave, K accumulated in 16 steps ---
        v8f acc = {};
        const int bbase = (s * NBATCH + bb) * REG_STRIDE;
        #pragma unroll
        for (int kc2 = 0; kc2 < 16; ++kc2) {
            const v2f a  = *(const v2f*)&lds[WWA_BASE + (((mt2 * 16 + kc2) * 32 + lane) << 1)];
            const v2f bm = *(const v2f*)&lds[bbase + kc2 * 64 + lane * 2];
            acc = __builtin_amdgcn_wmma_f32_16x16x4_f32(
                false, a, false, bm, (short)0, acc, false, false);
        }
        __syncthreads();   // all reads of region (s,*) and wwA complete

        // --- overwrite dead B region of head s with mixed[s] (layout m*16+v) ---
        const int mhi = (lane >> 4) << 3;
        const int vv  = lane & 15;
        #pragma unroll
        for (int j = 0; j < 8; ++j) {
            const int mm = mt2 * 16 + j + mhi;
            lds[bbase + mm * 16 + vv] = acc[j];
        }
    }
    __syncthreads();

    // ================= Epilogue: bias + gate, coalesced stream ===================
    for (int e = tid; e < NBATCH * NTOK * GATE; e += 512) {
        const int c = e & 255;
        const int n = (e >> 8) & 63;
        const int b = e >> 14;
        const int sH = c & 15, vv = c >> 4;
        const float mval = lds[(sH * NBATCH + b) * REG_STRIDE + n * 16 + vv];
        const float bias = tbias[w * NTOK + n];
        const size_t rowoff = ((size_t)b * NWIN + w) * NTOK + n;
        const float gate = x[rowoff * (2 * GATE) + GATE + c];
        out[rowoff * GATE + c] = (mval + bias) * gate;
    }
}

extern "C" void kernel_launch(void* const* d_in, const int* in_sizes, int n_in,
                              void* d_out, int out_size, void* d_ws, size_t ws_size,
                              hipStream_t stream) {
    (void)in_sizes; (void)n_in; (void)d_ws; (void)ws_size; (void)out_size;
    const float* x       = (const float*)d_in[0];
    const float* nw      = (const float*)d_in[1];
    const float* nb      = (const float*)d_in[2];
    const float* centers = (const float*)d_in[3];
    const float* spreads = (const float*)d_in[4];
    const float* tbias   = (const float*)d_in[5];
    float* out = (float*)d_out;

    const size_t shmem = (size_t)LDS_FLOATS * sizeof(float);   // ~281 KB of 320 KB WGP LDS
    sgating_fused_kernel<<<dim3(NWIN), dim3(512), shmem, stream>>>(
        x, nw, nb, centers, spreads, tbias, out);
}